// DeeperGCN_4509715661110
// MI455X (gfx1250) — compile-verified
//
#include <hip/hip_runtime.h>
#include <hip/hip_bf16.h>
#include <math.h>

typedef float v2f __attribute__((ext_vector_type(2)));
typedef float v8f __attribute__((ext_vector_type(8)));

#define NN    50000
#define HID_C 128
#define EPS_MSG 1e-7f

// ---------------------------------------------------------------------------
// FP32 WMMA GEMM:  C[M,N] = A[M,K] @ W[K,N] + bias  (+ optional residual)
// One wave (32 threads) computes one 16x16 output tile, K stepped by 4 using
// V_WMMA_F32_16X16X4_F32.  M, N assumed multiples of 16 (50000 = 3125*16).
// A-frag layout: lanes 0-15 M=0..15 / K=(0,1); lanes 16-31 M=0..15 / K=(2,3).
// B-frag layout: lanes hold column N=lane&15; K pairs split by lane half.
// D layout: VGPR v -> row v (lanes 0-15) / row v+8 (lanes 16-31), col = lane&15.
// ---------------------------------------------------------------------------
__global__ void gemm_wmma_f32(const float* __restrict__ A,
                              const float* __restrict__ W,
                              const float* __restrict__ bias,
                              const float* __restrict__ resid,
                              float* __restrict__ C,
                              int M, int K, int N, int add_resid)
{
    const int lane = threadIdx.x;        // 0..31, wave32
    const int half = lane >> 4;          // 0 or 1
    const int l15  = lane & 15;
    const int row  = blockIdx.x * 16;
    const int col  = blockIdx.y * 16;

    const float* Arow = A + (size_t)(row + l15) * K;

    v8f acc = {};
#pragma unroll 4
    for (int k = 0; k < K; k += 4) {
        const int kb = k + half * 2;
        v2f a, b;
        a.x = Arow[kb + 0];
        a.y = Arow[kb + 1];
        b.x = W[(size_t)(kb + 0) * N + col + l15];
        b.y = W[(size_t)(kb + 1) * N + col + l15];
        acc = __builtin_amdgcn_wmma_f32_16x16x4_f32(
                  false, a, false, b, (short)0, acc, false, false);
    }

    const float bv = bias[col + l15];
#pragma unroll
    for (int v = 0; v < 8; ++v) {
        const int r = row + v + half * 8;
        float val = acc[v] + bv;
        size_t off = (size_t)r * N + col + l15;
        if (add_resid) val += resid[off];
        C[off] = val;
    }
}

// ---------------------------------------------------------------------------
// Edge encoder: ea[E,128] = edge_attr[E,16] @ edge_W[16,128] + edge_b
// ---------------------------------------------------------------------------
__global__ void edge_encode(const float* __restrict__ EA,
                            const float* __restrict__ W,
                            const float* __restrict__ b,
                            float* __restrict__ ea, long long total)
{
    long long idx = (long long)blockIdx.x * blockDim.x + threadIdx.x;
    if (idx >= total) return;
    const int c = (int)(idx & (HID_C - 1));
    const long long e = idx >> 7;
    const float* row = EA + e * 16;
    float acc = b[c];
#pragma unroll
    for (int k = 0; k < 16; ++k)
        acc = fmaf(row[k], W[k * HID_C + c], acc);
    ea[idx] = acc;
}

// ---------------------------------------------------------------------------
// Row LayerNorm(+ReLU), one block per row, H threads.
// ---------------------------------------------------------------------------
template<int H>
__global__ void ln_act(const float* __restrict__ in,
                       const float* __restrict__ g,
                       const float* __restrict__ b,
                       float* __restrict__ out, int do_relu)
{
    __shared__ float red[H];
    const int n = blockIdx.x, c = threadIdx.x;
    const float v = in[(size_t)n * H + c];
    red[c] = v; __syncthreads();
    for (int st = H / 2; st > 0; st >>= 1) {
        if (c < st) red[c] += red[c + st];
        __syncthreads();
    }
    const float mu = red[0] * (1.0f / H);
    __syncthreads();
    const float xc = v - mu;
    red[c] = xc * xc; __syncthreads();
    for (int st = H / 2; st > 0; st >>= 1) {
        if (c < st) red[c] += red[c + st];
        __syncthreads();
    }
    const float var = red[0] * (1.0f / H);
    float y = xc * rsqrtf(var + 1e-5f) * g[c] + b[c];
    if (do_relu) y = fmaxf(y, 0.0f);
    out[(size_t)n * H + c] = y;
}

// ---------------------------------------------------------------------------
// Segment-softmax aggregation (2 edge passes + fixups)
// ---------------------------------------------------------------------------
__global__ void init_agg(float* __restrict__ m, float* __restrict__ s,
                         float* __restrict__ w, long long total)
{
    long long i = (long long)blockIdx.x * blockDim.x + threadIdx.x;
    if (i < total) { m[i] = -INFINITY; s[i] = 0.0f; w[i] = 0.0f; }
}

__device__ inline void atomicMaxF(float* addr, float val)
{
    if (val >= 0.0f) atomicMax((int*)addr, __float_as_int(val));
    else             atomicMin((unsigned int*)addr, __float_as_uint(val));
}

__global__ void edge_max(const float* __restrict__ xin,
                         const float* __restrict__ ea,
                         const int* __restrict__ src,
                         const int* __restrict__ dst,
                         const float* __restrict__ tptr,
                         long long total, float* __restrict__ m)
{
    long long idx = (long long)blockIdx.x * blockDim.x + threadIdx.x;
    if (idx >= total) return;
    const int c = (int)(idx & (HID_C - 1));
    const long long e = idx >> 7;
    const float t = tptr[0];
    const float msg = fmaxf(xin[(size_t)src[e] * HID_C + c] + ea[idx], 0.0f) + EPS_MSG;
    atomicMaxF(&m[(size_t)dst[e] * HID_C + c], msg * t);
}

__global__ void fix_m(float* __restrict__ m, long long total)
{
    long long i = (long long)blockIdx.x * blockDim.x + threadIdx.x;
    if (i < total) {
        const float v = m[i];
        m[i] = (v > -INFINITY) ? v : 0.0f;   // NaN / -inf -> 0 (isolated nodes)
    }
}

__global__ void edge_sum(const float* __restrict__ xin,
                         const float* __restrict__ ea,
                         const int* __restrict__ src,
                         const int* __restrict__ dst,
                         const float* __restrict__ tptr,
                         long long total,
                         const float* __restrict__ m,
                         float* __restrict__ s, float* __restrict__ w)
{
    long long idx = (long long)blockIdx.x * blockDim.x + threadIdx.x;
    if (idx >= total) return;
    const int c = (int)(idx & (HID_C - 1));
    const long long e = idx >> 7;
    const float t = tptr[0];
    const float msg = fmaxf(xin[(size_t)src[e] * HID_C + c] + ea[idx], 0.0f) + EPS_MSG;
    const size_t doff = (size_t)dst[e] * HID_C + c;
    const float ex = expf(msg * t - m[doff]);
    atomicAdd(&s[doff], ex);
    atomicAdd(&w[doff], ex * msg);
}

__global__ void node_out(const float* __restrict__ w, const float* __restrict__ s,
                         const float* __restrict__ xin, float* __restrict__ out,
                         long long total)
{
    long long i = (long long)blockIdx.x * blockDim.x + threadIdx.x;
    if (i < total) {
        const float ss = s[i];
        out[i] = w[i] / (ss > 0.0f ? ss : 1.0f) + xin[i];
    }
}

// ---------------------------------------------------------------------------
// Final LN/ReLU + graph-feature concat + regression head. One block per node.
// Graph-feature mapping is the reference's faithful .view sequence:
//   a = n / npg; q = a*GFD + c; f = q / G; g = q % G; extra[c] = gf[g*GFD + f]
// ---------------------------------------------------------------------------
__global__ void head_kernel(const float* __restrict__ x,
                            const float* __restrict__ ln_g,
                            const float* __restrict__ ln_b,
                            const float* __restrict__ gf,   // [G,2]
                            const float* __restrict__ W0,   // [130,128]
                            const float* __restrict__ b0,   // [128]
                            const float* __restrict__ W1,   // [128]
                            const float* __restrict__ b1,   // [1]
                            float* __restrict__ out,
                            int G, int npg)
{
    __shared__ float feat[HID_C + 2];
    __shared__ float red[HID_C];
    const int n = blockIdx.x, c = threadIdx.x;   // blockDim = 128

    // LayerNorm + ReLU over the 128 channels
    const float v = x[(size_t)n * HID_C + c];
    red[c] = v; __syncthreads();
    for (int st = HID_C / 2; st > 0; st >>= 1) {
        if (c < st) red[c] += red[c + st];
        __syncthreads();
    }
    const float mu = red[0] * (1.0f / HID_C);
    __syncthreads();
    const float xc = v - mu;
    red[c] = xc * xc; __syncthreads();
    for (int st = HID_C / 2; st > 0; st >>= 1) {
        if (c < st) red[c] += red[c + st];
        __syncthreads();
    }
    const float var = red[0] * (1.0f / HID_C);
    __syncthreads();
    feat[c] = fmaxf(xc * rsqrtf(var + 1e-5f) * ln_g[c] + ln_b[c], 0.0f);
    if (c < 2) {
        const int a = n / npg;
        const int q = a * 2 + c;
        feat[HID_C + c] = gf[(q % G) * 2 + (q / G)];
    }
    __syncthreads();

    // hidden = relu(feat @ W0 + b0), then out = hidden @ W1 + b1
    float h = b0[c];
#pragma unroll 2
    for (int k = 0; k < HID_C + 2; ++k)
        h = fmaf(feat[k], W0[k * HID_C + c], h);
    h = fmaxf(h, 0.0f);
    red[c] = h * W1[c]; __syncthreads();
    for (int st = HID_C / 2; st > 0; st >>= 1) {
        if (c < st) red[c] += red[c + st];
        __syncthreads();
    }
    if (c == 0) out[n] = red[0] + b1[0];
}

// ---------------------------------------------------------------------------
extern "C" void kernel_launch(void* const* d_in, const int* in_sizes, int n_in,
                              void* d_out, int out_size, void* d_ws, size_t ws_size,
                              hipStream_t stream)
{
    const float* X     = (const float*)d_in[0];   // [N,64]
    const int*   EI    = (const int*)  d_in[1];   // [2,E]
    const float* EAttr = (const float*)d_in[2];   // [E,16]
    const float* GF    = (const float*)d_in[3];   // [G,2]
    const float* nodeW = (const float*)d_in[4];
    const float* nodeB = (const float*)d_in[5];
    const float* edgeW = (const float*)d_in[6];
    const float* edgeB = (const float*)d_in[7];
    const float* ts    = (const float*)d_in[8];   // [4]
    const float* W1s   = (const float*)d_in[9];   // [4,128,256]
    const float* b1s   = (const float*)d_in[10];
    const float* g1s   = (const float*)d_in[11];
    const float* be1s  = (const float*)d_in[12];
    const float* W2s   = (const float*)d_in[13];  // [4,256,128]
    const float* b2s   = (const float*)d_in[14];
    const float* lngs  = (const float*)d_in[15];  // [4,128]
    const float* lnbs  = (const float*)d_in[16];
    const float* hW0   = (const float*)d_in[17];  // [130,128]
    const float* hb0   = (const float*)d_in[18];
    const float* hW1   = (const float*)d_in[19];  // [128,1]
    const float* hb1   = (const float*)d_in[20];

    const int N = in_sizes[0] / 64;               // 50000
    const long long E = in_sizes[2] / 16;         // 800000
    const int G = in_sizes[3] / 2;                // 50
    const int npg = N / G;
    const int* src = EI;
    const int* dst = EI + E;

    const long long NH = (long long)N * HID_C;    // node-channel count
    const long long EH = (long long)E * HID_C;    // edge-channel count

    // workspace carve-out
    char* ws = (char*)d_ws;
    float* x      = (float*)ws; ws += NH * sizeof(float);
    float* xin    = (float*)ws; ws += NH * sizeof(float);
    float* mbuf   = (float*)ws; ws += NH * sizeof(float);
    float* sbuf   = (float*)ws; ws += NH * sizeof(float);
    float* wbuf   = (float*)ws; ws += NH * sizeof(float);
    float* aggout = (float*)ws; ws += NH * sizeof(float);
    float* h1     = (float*)ws; ws += (long long)N * 256 * sizeof(float);
    float* ea     = (float*)ws; ws += EH * sizeof(float);

    const dim3 wave(32);
    const int nodeBlocks = (int)((NH + 255) / 256);
    const int edgeBlocks = (int)((EH + 255) / 256);

    // node encoder: x = X @ node_W + node_b   (WMMA f32, K=64)
    gemm_wmma_f32<<<dim3(N / 16, HID_C / 16), wave, 0, stream>>>(
        X, nodeW, nodeB, x, x, N, 64, HID_C, 0);

    // edge encoder: ea = edge_attr @ edge_W + edge_b
    edge_encode<<<edgeBlocks, 256, 0, stream>>>(EAttr, edgeW, edgeB, ea, EH);

    for (int l = 0; l < 4; ++l) {
        const float* cur = x;
        if (l > 0) {
            ln_act<HID_C><<<N, HID_C, 0, stream>>>(
                x, lngs + l * HID_C, lnbs + l * HID_C, xin, 1);
            cur = xin;
        }
        init_agg<<<nodeBlocks, 256, 0, stream>>>(mbuf, sbuf, wbuf, NH);
        edge_max<<<edgeBlocks, 256, 0, stream>>>(cur, ea, src, dst, ts + l, EH, mbuf);
        fix_m<<<nodeBlocks, 256, 0, stream>>>(mbuf, NH);
        edge_sum<<<edgeBlocks, 256, 0, stream>>>(cur, ea, src, dst, ts + l, EH,
                                                 mbuf, sbuf, wbuf);
        node_out<<<nodeBlocks, 256, 0, stream>>>(wbuf, sbuf, cur, aggout, NH);

        // MLP: h1 = LN(agg @ W1 + b1); relu; x (+)= h1 @ W2 + b2
        gemm_wmma_f32<<<dim3(N / 16, 256 / 16), wave, 0, stream>>>(
            aggout, W1s + (size_t)l * HID_C * 256, b1s + l * 256,
            h1, h1, N, HID_C, 256, 0);
        ln_act<256><<<N, 256, 0, stream>>>(
            h1, g1s + l * 256, be1s + l * 256, h1, 1);
        gemm_wmma_f32<<<dim3(N / 16, HID_C / 16), wave, 0, stream>>>(
            h1, W2s + (size_t)l * 256 * HID_C, b2s + l * HID_C,
            x, x, N, 256, HID_C, (l > 0) ? 1 : 0);
    }

    head_kernel<<<N, HID_C, 0, stream>>>(
        x, lngs, lnbs, GF, hW0, hb0, hW1, hb1, (float*)d_out, G, npg);
}